// GraphConvolutionSparse_46411416600779
// MI455X (gfx1250) — compile-verified
//
#include <hip/hip_runtime.h>
#include <hip/hip_bf16.h>

// ---------------------------------------------------------------------------
// GraphConvolutionSparse for MI455X (gfx1250, wave32)
//   xw  = densify(X_coo) @ W        (WMMA f32 16x16x4 GEMM, W staged in LDS)
//   out = relu( scatter_add_{edges}( adj_val * xw[adj_col] -> adj_row ) )
// ---------------------------------------------------------------------------

typedef __attribute__((ext_vector_type(2))) float v2f;
typedef __attribute__((ext_vector_type(8))) float v8f;

#define N_FEAT 256
#define OUT_DIM 128
#define KC 64                 // K-chunk staged in LDS: 64*128*4 = 32 KB

// ------------------------------ utility kernels ----------------------------

__global__ void zero_f32_kernel(float* __restrict__ p, long long n) {
    long long i = (long long)blockIdx.x * blockDim.x + threadIdx.x;
    if (i < n) p[i] = 0.0f;
}

__global__ void relu_inplace_kernel(float* __restrict__ p, long long n) {
    long long i = (long long)blockIdx.x * blockDim.x + threadIdx.x;
    if (i < n) p[i] = fmaxf(p[i], 0.0f);
}

// scatter COO feature nonzeros into dense Xd [n_nodes x 256]
__global__ void scatter_feat_kernel(const int* __restrict__ fr,
                                    const int* __restrict__ fc,
                                    const float* __restrict__ fv,
                                    float* __restrict__ Xd, int nnz) {
    int i = blockIdx.x * blockDim.x + threadIdx.x;
    if (i < nnz) {
        atomicAdd(&Xd[(size_t)fr[i] * N_FEAT + fc[i]], fv[i]);
    }
}

// ------------------------------ WMMA GEMM ----------------------------------
// xw[M x 128] = Xd[M x 256] * W[256 x 128], f32 accumulate.
// Block = 256 threads = 8 waves; wave w owns m-tile (blockIdx.x*8 + w).
// Each wave: 8 accumulators (16x16 tiles covering N=0..127), K stepped by 4.
__global__ void __launch_bounds__(256)
gemm_xw_wmma_kernel(const float* __restrict__ Xd,
                    const float* __restrict__ W,
                    float* __restrict__ xw, int M) {
    // LDS layout: element (kk, n) of chunk stored at ((kk>>1)*128 + n)*2 + (kk&1)
    // -> the two K-adjacent values a B-fragment needs are contiguous (ds_load_b64).
    __shared__ float Bs[KC * OUT_DIM];

    const int lane  = threadIdx.x & 31;
    const int wave  = threadIdx.x >> 5;
    const int mtile = blockIdx.x * 8 + wave;
    const bool active = (mtile * 16 + 15) < M;
    const int m0    = mtile * 16;
    const int nrow  = lane & 15;   // M index of A row / N index of B col / N of C
    const int khalf = lane >> 4;   // 0 -> K{0,1}, 1 -> K{2,3} within a k-step

    v8f acc[8] = {};

    for (int kb = 0; kb < N_FEAT; kb += KC) {
        __syncthreads();
        // cooperative load of W[kb..kb+KC) rows into LDS (pairs-interleaved)
        for (int q = threadIdx.x; q < (KC * OUT_DIM) / 4; q += 256) {
            int kk = q >> 5;              // 0..63
            int n0 = (q & 31) * 4;        // 0..124
            const float4 w4 = *(const float4*)(W + (size_t)(kb + kk) * OUT_DIM + n0);
            float* base = &Bs[(((kk >> 1) * OUT_DIM) + n0) * 2 + (kk & 1)];
            base[0] = w4.x; base[2] = w4.y; base[4] = w4.z; base[6] = w4.w;
        }
        __syncthreads();

        if (active) {
            const float* arow = Xd + (size_t)(m0 + nrow) * N_FEAT;
#pragma unroll 4
            for (int kl = 0; kl < KC; kl += 4) {
                // A fragment (16x4): this lane holds K = kl+2*khalf, kl+2*khalf+1
                const float2 a2 =
                    *(const float2*)(arow + kb + kl + 2 * khalf);
                v2f a; a.x = a2.x; a.y = a2.y;
                const int pairbase = ((kl + 2 * khalf) >> 1) * OUT_DIM;
#pragma unroll
                for (int t = 0; t < 8; ++t) {
                    const float2 b2 =
                        *(const float2*)&Bs[(pairbase + t * 16 + nrow) * 2];
                    v2f b; b.x = b2.x; b.y = b2.y;
                    acc[t] = __builtin_amdgcn_wmma_f32_16x16x4_f32(
                        false, a, false, b, (short)0, acc[t], false, false);
                }
            }
        }
    }

    if (active) {
        // C/D layout: vgpr r -> M = r + 8*khalf, N = nrow
        const int mbase = m0 + khalf * 8;
#pragma unroll
        for (int t = 0; t < 8; ++t) {
#pragma unroll
            for (int r = 0; r < 8; ++r) {
                xw[(size_t)(mbase + r) * OUT_DIM + t * 16 + nrow] = acc[t][r];
            }
        }
    }
}

// ------------------------- edge scatter (A @ xw) ---------------------------
// One wave per edge: lane gathers float4 of xw[col], atomically adds into row.
__global__ void __launch_bounds__(256)
spmm_edges_kernel(const int* __restrict__ ar, const int* __restrict__ ac,
                  const float* __restrict__ av,
                  const float* __restrict__ xw, float* __restrict__ out,
                  int nnz) {
    long long tid = (long long)blockIdx.x * blockDim.x + threadIdx.x;
    int e = (int)(tid >> 5);
    if (e >= nnz) return;
    const int lane = threadIdx.x & 31;
    const int r = ar[e];
    const int c = ac[e];
    const float v = av[e];
    const float4 x4 = ((const float4*)(xw + (size_t)c * OUT_DIM))[lane];
    float* op = out + (size_t)r * OUT_DIM + lane * 4;
    atomicAdd(op + 0, v * x4.x);
    atomicAdd(op + 1, v * x4.y);
    atomicAdd(op + 2, v * x4.z);
    atomicAdd(op + 3, v * x4.w);
}

// ---------------- fallback phase-1 (only if workspace is tiny) -------------
__global__ void __launch_bounds__(256)
spmm_feat_fallback_kernel(const int* __restrict__ fr, const int* __restrict__ fc,
                          const float* __restrict__ fv,
                          const float* __restrict__ W, float* __restrict__ xw,
                          int nnz) {
    long long tid = (long long)blockIdx.x * blockDim.x + threadIdx.x;
    int e = (int)(tid >> 5);
    if (e >= nnz) return;
    const int lane = threadIdx.x & 31;
    const int r = fr[e];
    const int c = fc[e];
    const float v = fv[e];
    const float4 w4 = ((const float4*)(W + (size_t)c * OUT_DIM))[lane];
    float* op = xw + (size_t)r * OUT_DIM + lane * 4;
    atomicAdd(op + 0, v * w4.x);
    atomicAdd(op + 1, v * w4.y);
    atomicAdd(op + 2, v * w4.z);
    atomicAdd(op + 3, v * w4.w);
}

// ------------------------------- launcher ----------------------------------

extern "C" void kernel_launch(void* const* d_in, const int* in_sizes, int n_in,
                              void* d_out, int out_size, void* d_ws, size_t ws_size,
                              hipStream_t stream) {
    const int*   feat_rows = (const int*)d_in[0];
    const int*   feat_cols = (const int*)d_in[1];
    const float* feat_vals = (const float*)d_in[2];
    const int*   adj_rows  = (const int*)d_in[3];
    const int*   adj_cols  = (const int*)d_in[4];
    const float* adj_vals  = (const float*)d_in[5];
    const float* W         = (const float*)d_in[6];

    const int nnz_x  = in_sizes[0];
    const int nnz_a  = in_sizes[3];
    const int n_nodes = out_size / OUT_DIM;   // avoids device read of d_in[7]

    float* out = (float*)d_out;
    const long long out_elems = (long long)n_nodes * OUT_DIM;
    const size_t xd_elems = (size_t)n_nodes * N_FEAT;
    const size_t xw_elems = (size_t)n_nodes * OUT_DIM;

    const bool wmma_path =
        ws_size >= (xd_elems + xw_elems) * sizeof(float) && (n_nodes % 16) == 0;

    float* Xd = (float*)d_ws;
    float* xw = wmma_path ? (Xd + xd_elems) : (float*)d_ws;

    // zero the output accumulator
    {
        long long n = out_elems;
        zero_f32_kernel<<<(unsigned)((n + 255) / 256), 256, 0, stream>>>(out, n);
    }

    if (wmma_path) {
        // zero dense X
        {
            long long n = (long long)xd_elems;
            zero_f32_kernel<<<(unsigned)((n + 255) / 256), 256, 0, stream>>>(Xd, n);
        }
        // densify X
        scatter_feat_kernel<<<(nnz_x + 255) / 256, 256, 0, stream>>>(
            feat_rows, feat_cols, feat_vals, Xd, nnz_x);
        // xw = Xd @ W via WMMA
        const int mtiles = n_nodes / 16;
        const unsigned gblocks = (unsigned)((mtiles + 7) / 8);
        gemm_xw_wmma_kernel<<<gblocks, 256, 0, stream>>>(Xd, W, xw, n_nodes);
    } else {
        // fallback: atomic scatter of W rows directly into xw
        {
            long long n = (long long)xw_elems;
            zero_f32_kernel<<<(unsigned)((n + 255) / 256), 256, 0, stream>>>(xw, n);
        }
        long long thr = (long long)nnz_x * 32;
        spmm_feat_fallback_kernel<<<(unsigned)((thr + 255) / 256), 256, 0, stream>>>(
            feat_rows, feat_cols, feat_vals, W, xw, nnz_x);
    }

    // out += A @ xw  (one wave per edge, 4 floats per lane)
    {
        long long thr = (long long)nnz_a * 32;
        spmm_edges_kernel<<<(unsigned)((thr + 255) / 256), 256, 0, stream>>>(
            adj_rows, adj_cols, adj_vals, xw, out, nnz_a);
    }

    // relu in place
    relu_inplace_kernel<<<(unsigned)((out_elems + 255) / 256), 256, 0, stream>>>(
        out, out_elems);
}